// GraphSAGE_59854664237965
// MI455X (gfx1250) — compile-verified
//
#include <hip/hip_runtime.h>

#define N_NODES 50000
#define NFEAT   64
#define N_TILES (N_NODES / 16)   // 3125, exact

typedef __attribute__((ext_vector_type(2))) float v2f;
typedef __attribute__((ext_vector_type(4))) float v4f;
typedef __attribute__((ext_vector_type(8))) float v8f;

// ---------------------------------------------------------------------------
// Zero a float buffer (float4 granularity; callers guarantee n % 4 == 0).
// ---------------------------------------------------------------------------
__global__ void zero_kernel(float* __restrict__ p, int n4) {
  v4f* p4 = (v4f*)p;
  int i = blockIdx.x * blockDim.x + threadIdx.x;
  const int stride = gridDim.x * blockDim.x;
  const v4f z = {0.f, 0.f, 0.f, 0.f};
  for (; i < n4; i += stride) p4[i] = z;
}

// Native fp32 atomic add (global_atomic_add_f32, RNE, no CAS loop).
__device__ __forceinline__ void fadd_atomic(float* p, float v) {
#if defined(__HIP_DEVICE_COMPILE__)
  unsafeAtomicAdd(p, v);
#else
  atomicAdd(p, v);
#endif
}

// ---------------------------------------------------------------------------
// Edge aggregation: one wave32 per edge. Lane l handles features 2l, 2l+1.
// agg[dst,:] += feat[src,:]; deg[dst] += 1 (first layer only).
// Edge index is wave-uniform: force it scalar so src/dst become s_loads.
// ---------------------------------------------------------------------------
__global__ __launch_bounds__(256) void edge_agg_kernel(
    const float* __restrict__ feat, const int* __restrict__ src,
    const int* __restrict__ dst, float* __restrict__ agg,
    float* __restrict__ deg, int n_edges, int add_deg)
{
  const int gid  = blockIdx.x * blockDim.x + threadIdx.x;
  const int e    = __builtin_amdgcn_readfirstlane(gid >> 5);  // wave-uniform
  if (e >= n_edges) return;
  const int lane = gid & 31;
  const int s = src[e];   // scalar loads (uniform address)
  const int d = dst[e];
  const v2f v = *(const v2f*)(feat + (size_t)s * NFEAT + lane * 2);
  float* ap = agg + (size_t)d * NFEAT + lane * 2;
  fadd_atomic(ap,     v.x);
  fadd_atomic(ap + 1, v.y);
  if (add_deg && lane == 0) fadd_atomic(deg + d, 1.0f);
}

// ---------------------------------------------------------------------------
// SAGE layer: out = act( hin @ Wself + (agg/deg) @ Wneigh + bias )
// 128 threads = 4 waves; wave w owns output columns [16w, 16w+16).
// Each 16x16 output tile = 32 chained V_WMMA_F32_16X16X4_F32 (K=64, 2 GEMMs).
// B fragments for both weight matrices live in VGPRs across the M loop.
// ---------------------------------------------------------------------------
__global__ __launch_bounds__(128) void sage_layer_kernel(
    const float* __restrict__ hin, const float* __restrict__ agg,
    const float* __restrict__ deg, const float* __restrict__ Wself,
    const float* __restrict__ Wneigh, const float* __restrict__ bias,
    float* __restrict__ out, int do_relu)
{
  const int lane = threadIdx.x & 31;
  const int wave = threadIdx.x >> 5;          // 0..3 -> N tile
  const int n0   = wave * 16;
  const int ncol = n0 + (lane & 15);          // this lane's output column
  const int koff = (lane >> 4) * 2;           // K sub-offset per A/B layout

  // Preload B fragments: B[k][n] with lane->n, (koff,koff+1) packed per v2f.
  v2f bs[16], bn[16];
#pragma unroll
  for (int kk = 0; kk < 16; ++kk) {
    const int k = kk * 4 + koff;
    v2f t;
    t.x = Wself[(k + 0) * NFEAT + ncol];
    t.y = Wself[(k + 1) * NFEAT + ncol];
    bs[kk] = t;
    t.x = Wneigh[(k + 0) * NFEAT + ncol];
    t.y = Wneigh[(k + 1) * NFEAT + ncol];
    bn[kk] = t;
  }
  const float bv = bias[ncol];

  for (int t = blockIdx.x; t < N_TILES; t += gridDim.x) {
    const int row0 = t * 16;
    const int m    = row0 + (lane & 15);      // A-matrix row for this lane
    const float invd = 1.0f / fmaxf(deg[m], 1.0f);
    const float* hrow = hin + (size_t)m * NFEAT;
    const float* arow = agg + (size_t)m * NFEAT;

    v8f c = {bv, bv, bv, bv, bv, bv, bv, bv};   // bias broadcast per column
#pragma unroll
    for (int kk = 0; kk < 16; ++kk) {
      const int k = kk * 4 + koff;
      v2f a;
      a.x = hrow[k];
      a.y = hrow[k + 1];
      c = __builtin_amdgcn_wmma_f32_16x16x4_f32(
              /*neg_a=*/false, a, /*neg_b=*/false, bs[kk],
              /*c_mod=*/(short)0, c, /*reuse_a=*/false, /*reuse_b=*/false);
      v2f g;
      g.x = arow[k]     * invd;
      g.y = arow[k + 1] * invd;
      c = __builtin_amdgcn_wmma_f32_16x16x4_f32(
              false, g, false, bn[kk], (short)0, c, false, false);
    }

#pragma unroll
    for (int r = 0; r < 8; ++r) {
      float v = c[r];
      if (do_relu) v = fmaxf(v, 0.0f);
      const int orow = row0 + r + (lane >> 4) * 8;   // C/D layout: M=r / r+8
      out[(size_t)orow * NFEAT + ncol] = v;
    }
  }
}

// ---------------------------------------------------------------------------
// Host-side orchestration (graph-capture safe: only kernel launches).
// ---------------------------------------------------------------------------
extern "C" void kernel_launch(void* const* d_in, const int* in_sizes, int n_in,
                              void* d_out, int out_size, void* d_ws, size_t ws_size,
                              hipStream_t stream) {
  const float* x   = (const float*)d_in[0];
  const int*   src = (const int*)d_in[1];
  const int*   dst = (const int*)d_in[2];
  const float* Ws1 = (const float*)d_in[3];
  const float* Wn1 = (const float*)d_in[4];
  const float* b1  = (const float*)d_in[5];
  const float* Ws2 = (const float*)d_in[6];
  const float* Wn2 = (const float*)d_in[7];
  const float* b2  = (const float*)d_in[8];
  float*       out = (float*)d_out;

  const int n_edges = in_sizes[1];

  // Workspace layout: agg[N*64] | deg[N] | h[N*64]  (~26 MB, all fp32)
  float* agg = (float*)d_ws;
  float* deg = agg + (size_t)N_NODES * NFEAT;
  float* h   = deg + N_NODES;

  const int zthreads = 256;
  const int edge_blocks  = (n_edges * 32 + 255) / 256;
  const int layer_blocks = 1024;

  // ---- Layer 1 ----
  {
    const int n4 = (N_NODES * NFEAT + N_NODES) / 4;   // agg + deg contiguous
    int zb = (n4 + zthreads - 1) / zthreads;
    if (zb > 2048) zb = 2048;
    zero_kernel<<<zb, zthreads, 0, stream>>>(agg, n4);
  }
  edge_agg_kernel<<<edge_blocks, 256, 0, stream>>>(x, src, dst, agg, deg,
                                                   n_edges, /*add_deg=*/1);
  sage_layer_kernel<<<layer_blocks, 128, 0, stream>>>(x, agg, deg, Ws1, Wn1, b1,
                                                      h, /*do_relu=*/1);

  // ---- Layer 2 (degree reused) ----
  {
    const int n4 = (N_NODES * NFEAT) / 4;
    int zb = (n4 + zthreads - 1) / zthreads;
    if (zb > 2048) zb = 2048;
    zero_kernel<<<zb, zthreads, 0, stream>>>(agg, n4);
  }
  edge_agg_kernel<<<edge_blocks, 256, 0, stream>>>(h, src, dst, agg, deg,
                                                   n_edges, /*add_deg=*/0);
  sage_layer_kernel<<<layer_blocks, 128, 0, stream>>>(h, agg, deg, Ws2, Wn2, b2,
                                                      out, /*do_relu=*/0);
}